// HEGN_Loss_75333726372154
// MI455X (gfx1250) — compile-verified
//
#include <hip/hip_runtime.h>
#include <hip/hip_bf16.h>

typedef __attribute__((ext_vector_type(2))) float v2f;
typedef __attribute__((ext_vector_type(8))) float v8f;

// Problem constants (match reference)
#define BB 8
#define NN 4096
#define MM 4096

// One wave handles a 16-row query tile against all target columns.
// Block = 256 threads = 8 waves => 128 query rows per block.
// gridDim.x = B * (Nq/128), gridDim.y = 2 (direction: 0 = x->y, 1 = y->x).
//
// Operand-folding trick: A rows hold (-2*x0, -2*x1, -2*x2, 1), B cols hold
// (y0, y1, y2, |y|^2). One v_wmma_f32_16x16x4_f32 then yields
//   P[n,m] = |y_m|^2 - 2*x_n.y_m
// and since |x_n|^2 is a per-row constant,
//   min_m d2[n,m] = |x_n|^2 + min_m P[n,m]
// so the per-iteration epilogue is only the running min.
__global__ void __launch_bounds__(256)
chamfer_min_kernel(const float* __restrict__ X,
                   const float* __restrict__ Y,
                   float* __restrict__ minx,   // [B*N] min over m of d2(x_n, y_m)
                   float* __restrict__ miny)   // [B*M] min over n of d2(x_n, y_m)
{
    const int dir = blockIdx.y;
    const float* __restrict__ Q = dir ? Y : X;       // queries  [B, Nq, 3]
    const float* __restrict__ T = dir ? X : Y;       // targets  [B, Mt, 3]
    float* __restrict__ out     = dir ? miny : minx; // [B, Nq]
    const int Nq = dir ? MM : NN;
    const int Mt = dir ? NN : MM;

    const int blocks_per_batch = Nq / 128;
    const int b      = blockIdx.x / blocks_per_batch;
    const int rowblk = blockIdx.x % blocks_per_batch;

    const int lane = threadIdx.x & 31;
    const int wid  = threadIdx.x >> 5;
    const int half = lane >> 4;     // 0: lanes 0-15 (K=0,1), 1: lanes 16-31 (K=2,3)
    const int l16  = lane & 15;

    const int rowbase = rowblk * 128 + wid * 16;     // query-row tile base within batch

    const float* __restrict__ q = Q + (size_t)b * Nq * 3;
    const float* __restrict__ t = T + (size_t)b * Mt * 3;

    // ---- A operand (resident): row n = (-2*x0, -2*x1, -2*x2, 1) ----
    const int arow = rowbase + l16;
    const float ax0 = q[arow * 3 + 0];
    const float ax1 = q[arow * 3 + 1];
    const float ax2 = q[arow * 3 + 2];
    v2f a;
    a.x = half ? (-2.0f * ax2) : (-2.0f * ax0);
    a.y = half ? 1.0f          : (-2.0f * ax1);

    // ---- per-row |x|^2, laid out to match C/D tile: VGPR r, this lane's half ----
    float xn[8];
#pragma unroll
    for (int r = 0; r < 8; ++r) {
        const int rr = rowbase + r + half * 8;
        const float x0 = q[rr * 3 + 0];
        const float x1 = q[rr * 3 + 1];
        const float x2 = q[rr * 3 + 2];
        xn[r] = x0 * x0 + x1 * x1 + x2 * x2;
    }

    float rmin[8];
#pragma unroll
    for (int r = 0; r < 8; ++r) rmin[r] = 3.402823466e38f;

    // ---- sweep all target columns, 16 at a time ----
#pragma unroll 4
    for (int ct = 0; ct < Mt; ct += 16) {
        const int col = ct + l16;
        const float y0 = t[col * 3 + 0];
        const float y1 = t[col * 3 + 1];
        const float y2 = t[col * 3 + 2];
        const float yn = fmaf(y2, y2, fmaf(y1, y1, y0 * y0));

        // unconditional speculative prefetch, 4 tiles ahead (invalid translations
        // are silently dropped per ISA prefetch rules); locality 3 -> near caches
        __builtin_prefetch(&t[(col + 64) * 3], 0, 3);

        // B operand: col m = (y0, y1, y2, |y|^2)
        v2f bb;
        bb.x = half ? y2 : y0;
        bb.y = half ? yn : y1;

        v8f c = {0.f, 0.f, 0.f, 0.f, 0.f, 0.f, 0.f, 0.f};
        // P = A(16x4) x B(4x16) : emits v_wmma_f32_16x16x4_f32
        c = __builtin_amdgcn_wmma_f32_16x16x4_f32(
                /*neg_a=*/false, a, /*neg_b=*/false, bb,
                /*c_mod=*/(short)0, c, /*reuse_a=*/false, /*reuse_b=*/false);

        // running row-min of P (xn added once at the end)
#pragma unroll
        for (int r = 0; r < 8; ++r) rmin[r] = fminf(rmin[r], c[r]);
    }

    // ---- reduce row-min across the 16 lanes of each half (xor stays in half),
    //      then add the per-row |x|^2 ----
#pragma unroll
    for (int r = 0; r < 8; ++r) {
        float v = rmin[r];
        v = fminf(v, __shfl_xor(v, 1));
        v = fminf(v, __shfl_xor(v, 2));
        v = fminf(v, __shfl_xor(v, 4));
        v = fminf(v, __shfl_xor(v, 8));
        rmin[r] = v + xn[r];
    }

    // lane 0 -> rows rowbase+0..7, lane 16 -> rows rowbase+8..15
    if (l16 == 0) {
        float* o = out + (size_t)b * Nq + rowbase + half * 8;
#pragma unroll
        for (int r = 0; r < 8; ++r) o[r] = rmin[r];
    }
}

// Single-block finalize: means of the min arrays + L_reg.
__global__ void __launch_bounds__(256)
finalize_kernel(const float* __restrict__ minx, const float* __restrict__ miny,
                const float* __restrict__ R,    const float* __restrict__ S,
                const float* __restrict__ t,    const float* __restrict__ Rgt,
                const float* __restrict__ Sgt,  const float* __restrict__ tgt,
                float* __restrict__ out)
{
    __shared__ float sbuf[256];
    const int tid = threadIdx.x;

    const float wx = 1.0f / (float)(BB * NN);
    const float wy = 1.0f / (float)(BB * MM);

    float s = 0.0f;
    for (int i = tid; i < BB * NN; i += 256) s += minx[i] * wx;
    for (int i = tid; i < BB * MM; i += 256) s += miny[i] * wy;
    sbuf[tid] = s;
    __syncthreads();

    for (int off = 128; off > 0; off >>= 1) {
        if (tid < off) sbuf[tid] += sbuf[tid + off];
        __syncthreads();
    }

    if (tid == 0) {
        float reg = 0.0f;
        for (int b = 0; b < BB; ++b) {
            const float* Rb = R   + b * 9;
            const float* Gb = Rgt + b * 9;
            for (int i = 0; i < 3; ++i) {
                for (int k = 0; k < 3; ++k) {
                    float acc = 0.0f;
                    for (int j = 0; j < 3; ++j) acc += Rb[i * 3 + j] * Gb[k * 3 + j];
                    acc -= (i == k) ? 1.0f : 0.0f;
                    reg += acc * acc;
                }
            }
            for (int j = 0; j < 3; ++j) {
                float d = S[b * 3 + j] - Sgt[b * 3 + j];
                reg += d * d;
                d = t[b * 3 + j] - tgt[b * 3 + j];   // t is [B,3,1] -> same flat layout
                reg += d * d;
            }
        }
        out[0] = sbuf[0] + reg;
    }
}

extern "C" void kernel_launch(void* const* d_in, const int* in_sizes, int n_in,
                              void* d_out, int out_size, void* d_ws, size_t ws_size,
                              hipStream_t stream) {
    const float* x   = (const float*)d_in[0];   // [B,N,3]
    const float* y   = (const float*)d_in[1];   // [B,M,3]
    const float* R   = (const float*)d_in[2];   // [B,3,3]
    const float* S   = (const float*)d_in[3];   // [B,3]
    const float* t   = (const float*)d_in[4];   // [B,3,1]
    const float* Rgt = (const float*)d_in[5];   // [B,3,3]
    const float* Sgt = (const float*)d_in[6];   // [B,3]
    const float* tgt = (const float*)d_in[7];   // [B,3]

    float* minx = (float*)d_ws;                 // B*N floats
    float* miny = minx + (size_t)BB * NN;       // B*M floats

    dim3 grid(BB * (NN / 128), 2, 1);
    chamfer_min_kernel<<<grid, 256, 0, stream>>>(x, y, minx, miny);
    finalize_kernel<<<1, 256, 0, stream>>>(minx, miny, R, S, t, Rgt, Sgt, tgt,
                                           (float*)d_out);
}